// PennylaneCircuit_65481071401219
// MI455X (gfx1250) — compile-verified
//
#include <hip/hip_runtime.h>
#include <hip/hip_bf16.h>

// ---------------------------------------------------------------------------
// 18-qubit statevector simulator for MI455X (gfx1250, wave32).
//
// Fusions:
//   * Rz*Ry*Rx per wire  -> one 2x2 complex matrix (prep kernel 1)
//   * 4-wire Kronecker   -> 16x16 complex matrix   (prep kernel 2)
//   * gate application   -> 16x16 complex GEMM on state tiles using
//                           v_wmma_f32_16x16x4_f32 (4 real GEMMs, 16 WMMAs/tile)
//   * 17-CNOT ladder     -> single suffix-XOR index permutation per layer
//
// State (2 MB complex64) + scratch live in d_ws; everything launches on
// `stream`; fully deterministic; no allocation / sync inside capture.
// ---------------------------------------------------------------------------

typedef float v2f __attribute__((ext_vector_type(2)));
typedef float v8f __attribute__((ext_vector_type(8)));

#define NQ 18
#define NSTATE (1u << NQ)          // 262144 amplitudes
#define NGROUPS 7                  // basis + 6 layers
#define NBLOCKS 5                  // 4-wire blocks: w=0,4,8,12 and embedded 14..17

__device__ __forceinline__ float2 cmul(float2 a, float2 b) {
    return make_float2(a.x * b.x - a.y * b.y, a.x * b.y + a.y * b.x);
}
__device__ __forceinline__ float2 cadd(float2 a, float2 b) {
    return make_float2(a.x + b.x, a.y + b.y);
}
__device__ __forceinline__ v2f mk2(float a, float b) {
    v2f v; v.x = a; v.y = b; return v;
}

// ---------------------------------------------------------------------------
// Prep 1: fuse Rz(az)*Ry(ay)*Rx(ax) into one 2x2 complex matrix per (group,wire)
// ---------------------------------------------------------------------------
__global__ void k_fuse(const float* __restrict__ params,
                       const float* __restrict__ basis,
                       float2* __restrict__ Uw) {
    int t = threadIdx.x;
    if (t >= NGROUPS * NQ) return;
    int g = t / NQ, w = t % NQ;
    float ax, ay, az;
    if (g == 0) {
        ax = basis[w * 3 + 0]; ay = basis[w * 3 + 1]; az = basis[w * 3 + 2];
    } else {
        const float* p = params + ((g - 1) * NQ + w) * 3;
        ax = p[0]; ay = p[1]; az = p[2];
    }
    float cx = cosf(ax * 0.5f), sx = sinf(ax * 0.5f);
    float2 rx[4] = { {cx, 0.f}, {0.f, -sx}, {0.f, -sx}, {cx, 0.f} };
    float cy = cosf(ay * 0.5f), sy = sinf(ay * 0.5f);
    float2 ry[4] = { {cy, 0.f}, {-sy, 0.f}, {sy, 0.f}, {cy, 0.f} };
    // m = Ry * Rx
    float2 m[4];
    m[0] = cadd(cmul(ry[0], rx[0]), cmul(ry[1], rx[2]));
    m[1] = cadd(cmul(ry[0], rx[1]), cmul(ry[1], rx[3]));
    m[2] = cadd(cmul(ry[2], rx[0]), cmul(ry[3], rx[2]));
    m[3] = cadd(cmul(ry[2], rx[1]), cmul(ry[3], rx[3]));
    // U = Rz * m  (Rz diagonal: e^{-i az/2}, e^{+i az/2})
    float cz = cosf(az * 0.5f), sz = sinf(az * 0.5f);
    float2 e0 = make_float2(cz, -sz), e1 = make_float2(cz, sz);
    Uw[t * 4 + 0] = cmul(e0, m[0]);
    Uw[t * 4 + 1] = cmul(e0, m[1]);
    Uw[t * 4 + 2] = cmul(e1, m[2]);
    Uw[t * 4 + 3] = cmul(e1, m[3]);
}

// ---------------------------------------------------------------------------
// Prep 2: Kronecker products -> 16x16 complex block matrices.
// Block b<4 covers wires 4b..4b+3; block 4 = I2 (x) I2 (x) U16 (x) U17 on
// wires 14..17 (identity on 14,15).
// ---------------------------------------------------------------------------
__global__ void k_kron(const float2* __restrict__ Uw, float2* __restrict__ Ublk) {
    int idx = blockIdx.x * blockDim.x + threadIdx.x;
    if (idx >= NGROUPS * NBLOCKS * 256) return;
    int g = idx / (NBLOCKS * 256);
    int rem = idx % (NBLOCKS * 256);
    int b = rem / 256, e = rem % 256;
    int r = e >> 4, c = e & 15;
    float2 v = make_float2(1.f, 0.f);
    #pragma unroll
    for (int j = 0; j < 4; j++) {
        int rb = (r >> (3 - j)) & 1, cb = (c >> (3 - j)) & 1;
        float2 f;
        if (b == 4 && j < 2) {
            f = (rb == cb) ? make_float2(1.f, 0.f) : make_float2(0.f, 0.f);
        } else {
            int w = (b == 4) ? (14 + j) : (4 * b + j);
            f = Uw[(g * NQ + w) * 4 + rb * 2 + cb];
        }
        v = cmul(v, f);
    }
    Ublk[idx] = v;
}

// ---------------------------------------------------------------------------
// Element index for (tile t, row m, col c) of a block gate at bit offset
// `shift` (inner size = 1<<shift). Tiles partition the state -> in-place safe.
// ---------------------------------------------------------------------------
__device__ __forceinline__ unsigned idx_elem(unsigned t, unsigned m, unsigned c,
                                             int shift) {
    if (shift >= 4) {
        unsigned tpo = 1u << (shift - 4);            // 16-col tiles per outer
        unsigned outer = t >> (shift - 4);
        unsigned innerBase = (t & (tpo - 1)) << 4;
        return (outer << (4 + shift)) + (m << shift) + innerBase + c;
    } else {
        unsigned opt = 16u >> shift;                 // outers per tile
        unsigned outer = t * opt + (c >> shift);
        unsigned inner = c & ((1u << shift) - 1);
        return (outer << (4 + shift)) + (m << shift) + inner;
    }
}

// ---------------------------------------------------------------------------
// Apply one 16x16 complex gate to the whole state via WMMA f32.
// One wave per 16x16 complex tile; 1024 tiles -> grid 128 x 256 (8 waves/blk).
// Dr = Ur*Sr + (-Ui)*Si ; Di = Ui*Sr + Ur*Si  -> 16 v_wmma_f32_16x16x4_f32.
// ---------------------------------------------------------------------------
__global__ void __launch_bounds__(256) k_gate(float2* __restrict__ st,
                                              const float2* __restrict__ U,
                                              int shift) {
    unsigned lane = threadIdx.x & 31;
    unsigned t = blockIdx.x * (blockDim.x >> 5) + (threadIdx.x >> 5);
    unsigned half = lane >> 4;      // 0: lanes 0-15, 1: lanes 16-31
    unsigned l = lane & 15;

    v2f Ar[4], Ai[4], Ain[4], Br[4], Bi[4];
    #pragma unroll
    for (int kc = 0; kc < 4; kc++) {
        // A (16x4): lane row M=l; VGPR0/1 hold K = 4kc+2*half, +1
        unsigned k0 = 4 * kc + 2 * half;
        float2 u0 = U[l * 16 + k0];
        float2 u1 = U[l * 16 + k0 + 1];
        Ar[kc]  = mk2(u0.x, u1.x);
        Ai[kc]  = mk2(u0.y, u1.y);
        Ain[kc] = mk2(-u0.y, -u1.y);
        // B (4x16): lane col N=l; VGPR0/1 hold K rows r0, r0+1
        unsigned r0 = 4 * kc + 2 * half;
        float2 s0 = st[idx_elem(t, r0,     l, shift)];
        float2 s1 = st[idx_elem(t, r0 + 1, l, shift)];
        Br[kc] = mk2(s0.x, s1.x);
        Bi[kc] = mk2(s0.y, s1.y);
    }

    v8f dr = {}; v8f di = {};
    #pragma unroll
    for (int kc = 0; kc < 4; kc++)
        dr = __builtin_amdgcn_wmma_f32_16x16x4_f32(false, Ar[kc], false, Br[kc],
                                                   (short)0, dr, false, false);
    #pragma unroll
    for (int kc = 0; kc < 4; kc++)
        dr = __builtin_amdgcn_wmma_f32_16x16x4_f32(false, Ain[kc], false, Bi[kc],
                                                   (short)0, dr, false, false);
    #pragma unroll
    for (int kc = 0; kc < 4; kc++)
        di = __builtin_amdgcn_wmma_f32_16x16x4_f32(false, Ai[kc], false, Br[kc],
                                                   (short)0, di, false, false);
    #pragma unroll
    for (int kc = 0; kc < 4; kc++)
        di = __builtin_amdgcn_wmma_f32_16x16x4_f32(false, Ar[kc], false, Bi[kc],
                                                   (short)0, di, false, false);

    // D (16x16 f32): lane col N=l; VGPR r holds row M = 8*half + r
    #pragma unroll
    for (int r = 0; r < 8; r++) {
        unsigned m = half * 8 + r;
        st[idx_elem(t, m, l, shift)] = make_float2(dr[r], di[r]);
    }
}

// ---------------------------------------------------------------------------
// Fused CNOT ladder (wires 0..16 -> targets 1..17): basis permutation
// out[suffixXor(n)] = in[n], suffixXor = MSB-to-LSB XOR scan.
// ---------------------------------------------------------------------------
__global__ void k_cnot(const float2* __restrict__ in, float2* __restrict__ out) {
    unsigned n = blockIdx.x * blockDim.x + threadIdx.x;
    unsigned y = n ^ (n >> 1);
    y ^= y >> 2; y ^= y >> 4; y ^= y >> 8; y ^= y >> 16;
    out[y] = in[n];
}

__global__ void k_init(float2* __restrict__ st) {
    unsigned n = blockIdx.x * blockDim.x + threadIdx.x;
    st[n] = (n == 0) ? make_float2(1.f, 0.f) : make_float2(0.f, 0.f);
}

// <Z0> = sum_n sign(bit17) * |amp|^2 : deterministic two-stage reduction.
__global__ void k_red1(const float2* __restrict__ st, float* __restrict__ partials) {
    __shared__ float sm[256];
    unsigned tid = blockIdx.x * 256 + threadIdx.x;
    float acc = 0.f;
    #pragma unroll
    for (int i = 0; i < 4; i++) {
        unsigned n = tid + i * 65536u;
        float2 a = st[n];
        float p = a.x * a.x + a.y * a.y;
        acc += (n & (1u << 17)) ? -p : p;
    }
    sm[threadIdx.x] = acc;
    __syncthreads();
    for (int s = 128; s > 0; s >>= 1) {
        if ((int)threadIdx.x < s) sm[threadIdx.x] += sm[threadIdx.x + s];
        __syncthreads();
    }
    if (threadIdx.x == 0) partials[blockIdx.x] = sm[0];
}

__global__ void k_red2(const float* __restrict__ partials, float* __restrict__ out) {
    __shared__ float sm[256];
    sm[threadIdx.x] = partials[threadIdx.x];
    __syncthreads();
    for (int s = 128; s > 0; s >>= 1) {
        if ((int)threadIdx.x < s) sm[threadIdx.x] += sm[threadIdx.x + s];
        __syncthreads();
    }
    if (threadIdx.x < 32) out[threadIdx.x] = sm[0];   // broadcast to batch of 32
}

// ---------------------------------------------------------------------------
extern "C" void kernel_launch(void* const* d_in, const int* in_sizes, int n_in,
                              void* d_out, int out_size, void* d_ws, size_t ws_size,
                              hipStream_t stream) {
    (void)in_sizes; (void)n_in; (void)out_size; (void)ws_size;
    // inputs: d_in[0] = inputs (unused by circuit), d_in[1] = params, d_in[2] = basis
    const float* params = (const float*)d_in[1];
    const float* basis  = (const float*)d_in[2];
    float* out = (float*)d_out;

    float2* ws   = (float2*)d_ws;
    float2* stA  = ws;                       // 2^18 complex (2 MB)
    float2* stB  = ws + NSTATE;              // 2^18 complex (2 MB)
    float2* Uw   = ws + 2 * NSTATE;          // 7*18 fused 2x2 (504 complex)
    float2* Ublk = Uw + NGROUPS * NQ * 4;    // 7*5 16x16 blocks (8960 complex)
    float*  partials = (float*)(Ublk + NGROUPS * NBLOCKS * 256); // 256 floats

    static const int shifts[NBLOCKS] = {14, 10, 6, 2, 0};

    k_init<<<NSTATE / 256, 256, 0, stream>>>(stA);
    k_fuse<<<1, 128, 0, stream>>>(params, basis, Uw);
    k_kron<<<(NGROUPS * NBLOCKS * 256 + 255) / 256, 256, 0, stream>>>(Uw, Ublk);

    float2* cur = stA;
    float2* oth = stB;
    for (int g = 0; g < NGROUPS; g++) {
        for (int b = 0; b < NBLOCKS; b++) {
            k_gate<<<128, 256, 0, stream>>>(cur, Ublk + (g * NBLOCKS + b) * 256,
                                            shifts[b]);
        }
        if (g >= 1) {   // groups 1..6 are layers followed by the CNOT ladder
            k_cnot<<<NSTATE / 256, 256, 0, stream>>>(cur, oth);
            float2* tmp = cur; cur = oth; oth = tmp;
        }
    }

    k_red1<<<256, 256, 0, stream>>>(cur, partials);
    k_red2<<<1, 256, 0, stream>>>(partials, out);
}